// GCN_77077483094029
// MI455X (gfx1250) — compile-verified
//
#include <hip/hip_runtime.h>
#include <math.h>

#define EMBED 32

typedef __attribute__((ext_vector_type(2))) float v2f;
typedef __attribute__((ext_vector_type(8))) float v8f;

// ---- Pass 1: deg = 1.0 (self loop), pooled = 0 ------------------------------
__global__ void k_init(float* __restrict__ deg, float* __restrict__ pooled,
                       int N, int GP) {
  int t = blockIdx.x * blockDim.x + threadIdx.x;
  if (t < N)  deg[t]    = 1.0f;
  if (t < GP) pooled[t] = 0.0f;
}

// ---- Pass 2: deg[dst] += 1 --------------------------------------------------
__global__ void k_degree(const int* __restrict__ dst, float* __restrict__ deg, int E) {
  int e = blockIdx.x * blockDim.x + threadIdx.x;
  if (e < E) atomicAdd(&deg[dst[e]], 1.0f);
}

// ---- Pass 3: agg[i] = W[x[i]] * dinv[i]^2 ; deg -> dinv in place ------------
// one wave32 per node, lane = feature; W staged in LDS
__global__ void k_embed(const int* __restrict__ x, const float* __restrict__ W,
                        float* __restrict__ dinv, float* __restrict__ agg,
                        int N, int ID) {
  extern __shared__ float sW[];
  for (int idx = threadIdx.x; idx < ID * EMBED; idx += blockDim.x) sW[idx] = W[idx];
  __syncthreads();
  int lane = threadIdx.x & 31;
  int i = (blockIdx.x * blockDim.x + threadIdx.x) >> 5;
  if (i >= N) return;
  int iu = __builtin_amdgcn_readfirstlane(i);   // wave-uniform -> scalar loads
  int   xi = x[iu];
  float r  = rsqrtf(dinv[iu]);                  // dinv[] currently holds deg
  agg[(size_t)i * EMBED + lane] = sW[xi * EMBED + lane] * (r * r);
  if (lane == 0) dinv[iu] = r;                  // overwrite deg with rsqrt(deg)
}

// ---- Pass 4: agg[dst] += dinv[src]*dinv[dst] * W[x[src]] --------------------
// one wave32 per edge, lane = feature; 32-lane atomic burst is contiguous 128B
__global__ void k_edge(const int* __restrict__ ei, const int* __restrict__ x,
                       const float* __restrict__ W, const float* __restrict__ dinv,
                       float* __restrict__ agg, int E, int ID) {
  extern __shared__ float sW[];
  for (int idx = threadIdx.x; idx < ID * EMBED; idx += blockDim.x) sW[idx] = W[idx];
  __syncthreads();
  int lane = threadIdx.x & 31;
  int e = (blockIdx.x * blockDim.x + threadIdx.x) >> 5;
  if (e >= E) return;
  int eu = __builtin_amdgcn_readfirstlane(e);   // wave-uniform edge id
  int   s    = ei[eu];
  int   d    = ei[E + eu];
  int   xs   = x[s];
  float norm = dinv[s] * dinv[d];
  float hv   = sW[xs * EMBED + lane];
  atomicAdd(&agg[(size_t)d * EMBED + lane], norm * hv);
}

// ---- Pass 5: pooled[batch[i]] += tanh(agg[i] + b_conv) ----------------------
__global__ void k_act_pool(const float* __restrict__ agg, const float* __restrict__ bconv,
                           const int* __restrict__ batch, float* __restrict__ pooled, int N) {
  int lane = threadIdx.x & 31;
  int i = (blockIdx.x * blockDim.x + threadIdx.x) >> 5;
  if (i >= N) return;
  int iu = __builtin_amdgcn_readfirstlane(i);
  int g  = batch[iu];
  float v = tanhf(agg[(size_t)i * EMBED + lane] + bconv[lane]);
  atomicAdd(&pooled[(size_t)g * EMBED + lane], v);
}

// ---- Pass 6: out[g] = pooled[g,:] @ W_out + b_out  via V_WMMA_F32_16X16X4_F32
// One wave handles 16 graphs (M=16). K=32 done as 8 chained K=4 WMMAs.
// A layout (fp32 16x4): lanes 0-15 hold M=lane {K=k0,k0+1}; lanes 16-31 hold
// M=lane-16 {K=k0+2,k0+3}. B is replicated identically across all N columns,
// so every column of D carries the same dot product; we read column 0 per the
// documented D layout (lane0: M=0..7 in v[0..7], lane16: M=8..15).
// Partial last tile: row index is CLAMPED (not zero-filled) — a clamped row
// only affects D rows that are never stored, so no divergent guarded loads.
__global__ void k_out_wmma(const float* __restrict__ pooled, const float* __restrict__ Wout,
                           const float* __restrict__ bout, float* __restrict__ out, int G) {
  int lane = threadIdx.x & 31;
  int wid  = (blockIdx.x * blockDim.x + threadIdx.x) >> 5;
  int g0   = wid * 16;
  if (g0 >= G) return;
  int M     = lane & 15;
  int khalf = (lane >> 4) * 2;              // 0 for lanes 0-15, 2 for 16-31
  int row   = g0 + M; if (row > G - 1) row = G - 1;   // clamp, always valid
  const float* prow = pooled + (size_t)row * EMBED + khalf;
  const float* pw   = Wout + khalf;

  v8f c = {};
#pragma unroll
  for (int k0 = 0; k0 < EMBED; k0 += 4) {
    v2f a = *(const v2f*)(prow + k0);       // 8B-aligned pair {K, K+1}
    v2f b = *(const v2f*)(pw + k0);
    c = __builtin_amdgcn_wmma_f32_16x16x4_f32(false, a, false, b,
                                              (short)0, c, false, false);
  }
  float bo = bout[0];
  if (lane == 0) {
#pragma unroll
    for (int r = 0; r < 8; ++r)
      if (g0 + r < G) out[g0 + r] = c[r] + bo;
  } else if (lane == 16) {
#pragma unroll
    for (int r = 0; r < 8; ++r)
      if (g0 + 8 + r < G) out[g0 + 8 + r] = c[r] + bo;
  }
}

extern "C" void kernel_launch(void* const* d_in, const int* in_sizes, int n_in,
                              void* d_out, int out_size, void* d_ws, size_t ws_size,
                              hipStream_t stream) {
  (void)n_in; (void)ws_size;
  const int*   x     = (const int*)d_in[0];
  const int*   ei    = (const int*)d_in[1];   // [2, E] row-major
  const int*   batch = (const int*)d_in[2];
  const float* W     = (const float*)d_in[3]; // [ID, 32]
  const float* bconv = (const float*)d_in[4]; // [32]
  const float* Wout  = (const float*)d_in[5]; // [32]
  const float* bout  = (const float*)d_in[6]; // [1]
  float*       out   = (float*)d_out;         // [G]

  const int N  = in_sizes[0];
  const int E  = in_sizes[1] / 2;
  const int G  = out_size;
  const int ID = in_sizes[3] / EMBED;         // 65

  // workspace carve: dinv[N] | agg[N*32] | pooled[G*32]  (~69 MB)
  float* dinv   = (float*)d_ws;
  float* agg    = dinv + N;
  float* pooled = agg + (size_t)N * EMBED;

  const int B = 256;
  const size_t shmem = (size_t)ID * EMBED * sizeof(float);   // 8320 B

  int initMax = (N > G * EMBED) ? N : G * EMBED;
  k_init<<<(initMax + B - 1) / B, B, 0, stream>>>(dinv, pooled, N, G * EMBED);

  k_degree<<<(E + B - 1) / B, B, 0, stream>>>(ei + E, dinv, E);

  long long nThr = (long long)N * 32;
  k_embed<<<(int)((nThr + B - 1) / B), B, shmem, stream>>>(x, W, dinv, agg, N, ID);

  long long eThr = (long long)E * 32;
  k_edge<<<(int)((eThr + B - 1) / B), B, shmem, stream>>>(ei, x, W, dinv, agg, E, ID);

  k_act_pool<<<(int)((nThr + B - 1) / B), B, 0, stream>>>(agg, bconv, batch, pooled, N);

  int waves  = (G + 15) / 16;
  int oThr   = waves * 32;
  k_out_wmma<<<(oThr + B - 1) / B, B, 0, stream>>>(pooled, Wout, bout, out, G);
}